// GateLoopedAttention_53291954209072
// MI455X (gfx1250) — compile-verified
//
#include <hip/hip_runtime.h>
#include <hip/hip_bf16.h>
#include <math.h>

// ---------------------------------------------------------------------------
// GateLoopedAttention on MI455X (gfx1250), fp32 end-to-end via
// V_WMMA_F32_16X16X4_F32.  Problem: B=2, N=2048, DIM=512, DI=128.
// ~4 GFLOP total -> bandwidth/latency bound.  GEMMs ride wave32 WMMA; the
// sequential scan double-buffers its per-chunk operands in LDS using CDNA5
// async global->LDS copies (ASYNCcnt) so HBM latency overlaps WMMA work.
// ---------------------------------------------------------------------------

typedef __attribute__((ext_vector_type(2))) float v2f;
typedef __attribute__((ext_vector_type(8))) float v8f;

#define DIM   512
#define DI    128
#define BATCH 2
#define SEQ   2048
#define NTOK  (BATCH * SEQ)   // 4096
#define CHUNK 16
#define NCHUNK (SEQ / CHUNK)  // 128

static __device__ __forceinline__
v8f wmma_f32(v2f a, v2f b, v8f c) {
  // D = A(16x4, f32) * B(4x16, f32) + C(16x16, f32)
  return __builtin_amdgcn_wmma_f32_16x16x4_f32(
      /*neg_a=*/false, a, /*neg_b=*/false, b,
      /*c_mod=*/(short)0, c, /*reuse_a=*/false, /*reuse_b=*/false);
}

// LDS byte address of a __shared__ object (the low 32 bits of the generic pointer are the LDS address)
static __device__ __forceinline__ unsigned lds_addr(const void* p) {
  return (unsigned)(uintptr_t)p;
}

// GLOBAL_LOAD_ASYNC_TO_LDS_B128: copies 16 bytes from global address vaddr to LDS address vdst, tracked by ASYNCcnt.
static __device__ __forceinline__ void async_b128(unsigned lds, const void* g) {
  asm volatile("global_load_async_to_lds_b128 %0, %1, off"
               :: "v"(lds), "v"((unsigned long long)(uintptr_t)g)
               : "memory");
}
static __device__ __forceinline__ void wait_async0() {
  asm volatile("s_wait_asynccnt 0x0" ::: "memory");
}

// ---------------------------------------------------------------------------
// Kernel 1: RMSNorm.  One wave per token; lane covers 4 float4s strided by 128 floats.
// xn = x / max(||x||,1e-12) * sqrt(512) * gamma
// ---------------------------------------------------------------------------
__global__ __launch_bounds__(256) void rmsnorm_kernel(
    const float* __restrict__ x, const float* __restrict__ gamma,
    float* __restrict__ xn) {
  const int token = blockIdx.x * (blockDim.x >> 5) + (threadIdx.x >> 5);
  const int lane  = threadIdx.x & 31;
  const float* xr = x + (size_t)token * DIM;
  float*       yr = xn + (size_t)token * DIM;

  float4 v[4];
  float ss = 0.f;
#pragma unroll
  for (int i = 0; i < 4; ++i) {
    v[i] = ((const float4*)xr)[lane + 32 * i];
    ss += v[i].x * v[i].x + v[i].y * v[i].y + v[i].z * v[i].z + v[i].w * v[i].w;
  }
#pragma unroll
  for (int off = 16; off > 0; off >>= 1) ss += __shfl_xor(ss, off, 32);

  const float nrm = fmaxf(sqrtf(ss), 1e-12f);
  const float s   = 22.62741699796952f / nrm;  // sqrt(512)/||x||
#pragma unroll
  for (int i = 0; i < 4; ++i) {
    const float4 g4 = ((const float4*)gamma)[lane + 32 * i];
    float4 o;
    o.x = v[i].x * s * g4.x;  o.y = v[i].y * s * g4.y;
    o.z = v[i].z * s * g4.z;  o.w = v[i].w * s * g4.w;
    ((float4*)yr)[lane + 32 * i] = o;
  }
}

// ---------------------------------------------------------------------------
// Kernel 2: generic row-major fp32 WMMA GEMM  C[M,N] = A[M,K] @ B[K,N].
// Block = 128 threads (4 waves).  A 16xK tile staged in LDS (contiguous,
// float4 coalesced); each wave owns one 16-wide column tile.  Weights (B)
// are tiny and L2-resident across the 256 row blocks.
// M%16==0, N%64==0, K%4==0 required.
// ---------------------------------------------------------------------------
__global__ __launch_bounds__(128) void gemm16_kernel(
    const float* __restrict__ A, const float* __restrict__ B,
    float* __restrict__ C, int M, int N, int K) {
  __shared__ float sA[16 * DIM];  // up to K=512
  const int tid  = threadIdx.x;
  const int wave = tid >> 5;
  const int lane = tid & 31;
  const int m    = lane & 15;
  const int hi   = lane >> 4;

  const float* Ab = A + (size_t)blockIdx.x * 16 * K;
  const int n4 = (16 * K) >> 2;
  for (int i = tid; i < n4; i += 128)
    ((float4*)sA)[i] = ((const float4*)Ab)[i];
  __syncthreads();

  const int n0 = (blockIdx.y * 4 + wave) * 16;
  v8f acc = {};
  for (int k0 = 0; k0 < K; k0 += 4) {
    const int ka = k0 + 2 * hi;
    v2f a = { sA[m * K + ka], sA[m * K + ka + 1] };
    v2f b = { B[(size_t)ka * N + n0 + m], B[(size_t)(ka + 1) * N + n0 + m] };
    acc = wmma_f32(a, b, acc);
  }
  float* Cb = C + (size_t)blockIdx.x * 16 * N + n0;
#pragma unroll
  for (int r = 0; r < 8; ++r) Cb[(size_t)(r + 8 * hi) * N + m] = acc[r];
}

// ---------------------------------------------------------------------------
// Kernel 3: gate.  ap[t, 2c]=real, ap[t,2c+1]=imag (+ba).
// g = Re( sigmoid(|a|) * e^{i*angle(a)} ) = sigmoid(r) * re/r
// ---------------------------------------------------------------------------
__global__ __launch_bounds__(256) void gate_kernel(
    const float* __restrict__ ap, const float* __restrict__ ba,
    float* __restrict__ g) {
  const int idx = blockIdx.x * blockDim.x + threadIdx.x;
  if (idx >= NTOK * DI) return;
  const int t = idx >> 7, c = idx & (DI - 1);
  const float xr = ap[(size_t)t * (2 * DI) + 2 * c]     + ba[2 * c];
  const float xi = ap[(size_t)t * (2 * DI) + 2 * c + 1] + ba[2 * c + 1];
  const float r  = sqrtf(xr * xr + xi * xi);
  const float sg = 1.0f / (1.0f + __expf(-r));
  const float co = (r > 1e-20f) ? (xr / r) : 1.0f;
  g[idx] = sg * co;
}

// ---------------------------------------------------------------------------
// Kernel 4: chunked GLA scan.  grid = 8 (batch x 4 e-tiles of 32), 1 wave.
// Raw q/k/g/v chunk tiles are double-buffered in LDS via async global->LDS
// B128 copies: chunk i+1 streams in (52 issues/lane, ~26KB) while chunk i's
// cumprod prep + ~168 WMMAs execute; one s_wait_asynccnt 0 per chunk.
// Per chunk (16 tokens):
//   A_i[d]  = prod_{t<=i} g_t[d]   (magnitude-clamped cumprod)
//   q~ = q*A ; k~ = k/A ; k^ = k*Atot/A
//   S   = tril(q~ k~^T)                       (WMMA, K=128)
//   out = q~ @ state + S @ V                  (WMMA, K=128 and K=16)
//   state = Atot (.) state + k^T @ V          (WMMA, K=16, 8 d-tiles)
// ---------------------------------------------------------------------------
__global__ __launch_bounds__(32) void scan_kernel(
    const float* __restrict__ qkv,  // [NTOK, 384] : q|k|v
    const float* __restrict__ g,    // [NTOK, 128]
    float* __restrict__ aout) {     // [NTOK, 128]
  const int b  = blockIdx.x >> 2;
  const int e0 = (blockIdx.x & 3) * 32;
  const int lane = threadIdx.x;
  const int m  = lane & 15;
  const int hi = lane >> 4;

  __shared__ float sG[2][CHUNK * DI];   // raw gates     (async dest)
  __shared__ float sQ[2][CHUNK * DI];   // raw q         (async dest)
  __shared__ float sK[2][CHUNK * DI];   // raw k         (async dest)
  __shared__ float sV[2][CHUNK][32];    // raw v e-slice (async dest)
  __shared__ float sQt[CHUNK][DI];      // q~   (token x d)
  __shared__ float sKr[CHUNK][DI];      // k~   (token x d)
  __shared__ float sKh[DI][CHUNK];      // k^T  (d x token)
  __shared__ float sS [CHUNK][CHUNK];
  __shared__ float sAtot[DI];
  __shared__ float sState[DI][32];

  for (int i = lane; i < DI * 32; i += 32) (&sState[0][0])[i] = 0.f;
  __syncthreads();

  const float* qkvb = qkv  + (size_t)b * SEQ * (3 * DI);
  const float* gb   = g    + (size_t)b * SEQ * DI;
  float*       ob   = aout + (size_t)b * SEQ * DI;

  // --- async stage of one chunk's raw tiles into LDS buffer `buf` ---------
  auto stage = [&](int buf, int t0) {
    const unsigned lG = lds_addr(&sG[buf][0]);
    const unsigned lQ = lds_addr(&sQ[buf][0]);
    const unsigned lK = lds_addr(&sK[buf][0]);
    const unsigned lV = lds_addr(&sV[buf][0][0]);
    // gates: 16x128 contiguous (8KB)
    const char* gsrc = (const char*)(gb + (size_t)t0 * DI);
#pragma unroll
    for (int j = 0; j < 16; ++j) {
      const unsigned off = (unsigned)(lane + 32 * j) * 16u;
      async_b128(lG + off, gsrc + off);
    }
    // q,k: 512B row slices, row stride 1536B
#pragma unroll
    for (int j = 0; j < 16; ++j) {
      const char* row = (const char*)(qkvb + (size_t)(t0 + j) * (3 * DI));
      async_b128(lQ + (unsigned)(j * 512 + lane * 16), row + lane * 16);
      async_b128(lK + (unsigned)(j * 512 + lane * 16), row + 512 + lane * 16);
    }
    // v e-slice: 16 rows x 128B
#pragma unroll
    for (int j = 0; j < 4; ++j) {
      const int idx = lane + 32 * j;
      const int row = idx >> 3, piece = idx & 7;
      const char* vsrc =
          (const char*)(qkvb + (size_t)(t0 + row) * (3 * DI) + 2 * DI + e0);
      async_b128(lV + (unsigned)(row * 128 + piece * 16), vsrc + piece * 16);
    }
  };

  stage(0, 0);  // prologue: chunk 0

  for (int ch = 0; ch < NCHUNK; ++ch) {
    const int t0  = ch * CHUNK;
    const int cur = ch & 1;

    wait_async0();      // chunk ch's tiles resident in LDS
    __syncthreads();
    if (ch + 1 < NCHUNK) stage(cur ^ 1, t0 + CHUNK);  // overlap next chunk

    // ---- prep: cumprods + scaled q/k, each lane owns 4 d-channels -------
#pragma unroll
    for (int dd = 0; dd < 4; ++dd) {
      const int d = lane + 32 * dd;
      float P = 1.f;
      float kr[CHUNK];
#pragma unroll
      for (int t = 0; t < CHUNK; ++t) {
        P *= sG[cur][t * DI + d];
        if (fabsf(P) < 1e-20f) P = (P >= 0.f) ? 1e-20f : -1e-20f;  // stability
        const float qv = sQ[cur][t * DI + d];
        const float kv = sK[cur][t * DI + d];
        sQt[t][d] = qv * P;
        kr[t] = kv / P;
        sKr[t][d] = kr[t];
      }
      sAtot[d] = P;
#pragma unroll
      for (int t = 0; t < CHUNK; ++t) sKh[d][t] = kr[t] * P;
    }
    __syncthreads();

    // ---- S = tril(q~ k~^T), K = 128 -------------------------------------
    v8f accS = {};
    for (int k0 = 0; k0 < DI; k0 += 4) {
      const int ka = k0 + 2 * hi;
      v2f a = { sQt[m][ka], sQt[m][ka + 1] };
      v2f bb = { sKr[m][ka], sKr[m][ka + 1] };   // B[K=d][N=token m]
      accS = wmma_f32(a, bb, accS);
    }
#pragma unroll
    for (int r = 0; r < 8; ++r) {
      const int row = r + 8 * hi;
      sS[row][m] = (m <= row) ? accS[r] : 0.f;   // causal mask
    }
    __syncthreads();

    // ---- out = q~ @ state + S @ V ---------------------------------------
#pragma unroll
    for (int nt = 0; nt < 2; ++nt) {
      v8f acc = {};
      for (int k0 = 0; k0 < DI; k0 += 4) {
        const int ka = k0 + 2 * hi;
        v2f a = { sQt[m][ka], sQt[m][ka + 1] };
        v2f bb = { sState[ka][nt * 16 + m], sState[ka + 1][nt * 16 + m] };
        acc = wmma_f32(a, bb, acc);
      }
#pragma unroll
      for (int k0 = 0; k0 < CHUNK; k0 += 4) {
        const int ka = k0 + 2 * hi;
        v2f a = { sS[m][ka], sS[m][ka + 1] };
        v2f bb = { sV[cur][ka][nt * 16 + m], sV[cur][ka + 1][nt * 16 + m] };
        acc = wmma_f32(a, bb, acc);
      }
#pragma unroll
      for (int r = 0; r < 8; ++r)
        ob[(size_t)(t0 + r + 8 * hi) * DI + e0 + nt * 16 + m] = acc[r];
    }
    __syncthreads();  // all reads of sState done before updates

    // ---- state = Atot (.) state + k^T @ V -------------------------------
#pragma unroll
    for (int dt = 0; dt < 8; ++dt) {
#pragma unroll
      for (int nt = 0; nt < 2; ++nt) {
        v8f c;
#pragma unroll
        for (int r = 0; r < 8; ++r) {
          const int drow = dt * 16 + r + 8 * hi;
          c[r] = sState[drow][nt * 16 + m] * sAtot[drow];
        }
#pragma unroll
        for (int k0 = 0; k0 < CHUNK; k0 += 4) {
          const int ka = k0 + 2 * hi;
          v2f a = { sKh[dt * 16 + m][ka], sKh[dt * 16 + m][ka + 1] };
          v2f bb = { sV[cur][ka][nt * 16 + m], sV[cur][ka + 1][nt * 16 + m] };
          c = wmma_f32(a, bb, c);
        }
#pragma unroll
        for (int r = 0; r < 8; ++r)
          sState[dt * 16 + r + 8 * hi][nt * 16 + m] = c[r];
      }
    }
    __syncthreads();
  }
}

// ---------------------------------------------------------------------------
extern "C" void kernel_launch(void* const* d_in, const int* in_sizes, int n_in,
                              void* d_out, int out_size, void* d_ws, size_t ws_size,
                              hipStream_t stream) {
  const float* x     = (const float*)d_in[0];  // [2,2048,512]
  const float* gamma = (const float*)d_in[1];  // [512]
  const float* Wqkv  = (const float*)d_in[2];  // [512,384]
  const float* Wa    = (const float*)d_in[3];  // [512,256]
  const float* ba    = (const float*)d_in[4];  // [256]
  const float* Wout  = (const float*)d_in[5];  // [128,512]
  float* out = (float*)d_out;                  // [2,2048,512] fp32

  float* ws   = (float*)d_ws;
  float* xn   = ws;                               // 4096*512
  float* qkv  = xn   + (size_t)NTOK * DIM;        // 4096*384
  float* ap   = qkv  + (size_t)NTOK * 3 * DI;     // 4096*256
  float* gbuf = ap   + (size_t)NTOK * 2 * DI;     // 4096*128
  float* aout = gbuf + (size_t)NTOK * DI;         // 4096*128

  // 1) RMSNorm (one wave/token, 8 waves/block)
  rmsnorm_kernel<<<NTOK / 8, 256, 0, stream>>>(x, gamma, xn);
  // 2) qkv = xn @ Wqkv  [4096,512]@[512,384]
  gemm16_kernel<<<dim3(NTOK / 16, (3 * DI) / 64), 128, 0, stream>>>(
      xn, Wqkv, qkv, NTOK, 3 * DI, DIM);
  // 3) ap = xn @ Wa     [4096,512]@[512,256]
  gemm16_kernel<<<dim3(NTOK / 16, (2 * DI) / 64), 128, 0, stream>>>(
      xn, Wa, ap, NTOK, 2 * DI, DIM);
  // 4) gate g = Re(polar(sigmoid|a|, angle a))
  gate_kernel<<<(NTOK * DI + 255) / 256, 256, 0, stream>>>(ap, ba, gbuf);
  // 5) chunked GLA scan: 2 batches x 4 e-tiles, async double-buffered
  scan_kernel<<<BATCH * 4, 32, 0, stream>>>(qkv, gbuf, aout);
  // 6) out = aout @ Wout [4096,128]@[128,512]
  gemm16_kernel<<<dim3(NTOK / 16, DIM / 64), 128, 0, stream>>>(
      aout, Wout, out, NTOK, DIM, DI);
}